// GATMultitask_36799279792418
// MI455X (gfx1250) — compile-verified
//
#include <hip/hip_runtime.h>
#include <hip/hip_bf16.h>
#include <math.h>

#define N_NODES 100000
#define N_EDGES 1600000
#define E_TOT   (N_EDGES + N_NODES)   // self loops appended
#define IN_CH   128
#define HID     32
#define HEADS   4
#define HIDTOT  (HEADS * HID)         // 128

typedef __attribute__((ext_vector_type(2))) float v2f;
typedef __attribute__((ext_vector_type(8))) float v8f;

static __device__ __forceinline__ float lrelu(float v) {
  return v > 0.0f ? v : 0.2f * v;
}

// Float atomic max via sign-split int atomics (monotone in both domains,
// correct for -0.0 and -inf init).
static __device__ __forceinline__ void atomicMaxF32(float* addr, float val) {
  int bits = __float_as_int(val);
  if (bits >= 0) atomicMax((int*)addr, bits);
  else           atomicMin((unsigned int*)addr, (unsigned int)bits);
}

__global__ void fill_f32(float* __restrict__ p, float v, int n) {
  int i = blockIdx.x * blockDim.x + threadIdx.x;
  if (i < n) p[i] = v;
}

// C[M,N] = A[M,K] x B[K,N], row-major fp32, one wave per 16x16 tile,
// v_wmma_f32_16x16x4_f32 over K in steps of 4. M,N multiples of 16, K of 4.
__global__ void gemm_f32_wmma(const float* __restrict__ A,
                              const float* __restrict__ B,
                              float* __restrict__ C, int M, int N, int K) {
  int wave = (blockIdx.x * blockDim.x + threadIdx.x) >> 5;
  int lane = threadIdx.x & 31;
  int ntiles = N >> 4;
  int mt = wave / ntiles;
  int nt = wave - mt * ntiles;
  if (mt * 16 >= M) return;                 // wave-uniform guard
  int l16 = lane & 15;
  int hi  = lane >> 4;                      // K-half select (0 or 1)
  const float* arow = A + (size_t)(mt * 16 + l16) * K + hi * 2;
  const float* bcol = B + (size_t)(hi * 2) * N + nt * 16 + l16;
  v8f acc = {};
  for (int k = 0; k < K; k += 4) {
    v2f a, b;
    a.x = arow[k];
    a.y = arow[k + 1];
    b.x = bcol[(size_t)k * N];
    b.y = bcol[(size_t)(k + 1) * N];
#if defined(__gfx1250__)
    acc = __builtin_amdgcn_wmma_f32_16x16x4_f32(
        /*neg_a=*/false, a, /*neg_b=*/false, b,
        /*c_mod=*/(short)0, acc, /*reuse_a=*/false, /*reuse_b=*/false);
#else
    acc[0] += a.x * b.x + a.y * b.y;        // host/other-arch pass placeholder
#endif
  }
  float* crow = C + (size_t)(mt * 16 + hi * 8) * N + nt * 16 + l16;
#pragma unroll
  for (int r = 0; r < 8; ++r) crow[(size_t)r * N] = acc[r];
}

// Per-node attention logits, layer 1: i = n*HEADS + h
__global__ void s1_kernel(const float* __restrict__ h1,
                          const float* __restrict__ as1,
                          const float* __restrict__ ad1,
                          float* __restrict__ s1s, float* __restrict__ s1d) {
  int i = blockIdx.x * blockDim.x + threadIdx.x;
  if (i >= N_NODES * HEADS) return;
  int n = i >> 2, h = i & 3;
  const float* hp  = h1 + (size_t)n * HIDTOT + h * HID;
  const float* asp = as1 + h * HID;
  const float* adp = ad1 + h * HID;
  float ss = 0.f, sd = 0.f;
#pragma unroll
  for (int d = 0; d < HID; ++d) { float v = hp[d]; ss += v * asp[d]; sd += v * adp[d]; }
  s1s[i] = ss; s1d[i] = sd;
}

static __device__ __forceinline__ void edge_sd(int e, const int* __restrict__ ei,
                                               int& src, int& dst) {
  if (e < N_EDGES) { src = ei[e]; dst = ei[N_EDGES + e]; }
  else             { src = e - N_EDGES; dst = src; }
}

__global__ void edge_max1(const int* __restrict__ ei, const float* __restrict__ s1s,
                          const float* __restrict__ s1d, float* __restrict__ m1) {
  int e = blockIdx.x * blockDim.x + threadIdx.x;
  if (e >= E_TOT) return;
  int src, dst; edge_sd(e, ei, src, dst);
#pragma unroll
  for (int h = 0; h < HEADS; ++h) {
    float v = lrelu(s1s[src * HEADS + h] + s1d[dst * HEADS + h]);
    atomicMaxF32(&m1[dst * HEADS + h], v);
  }
}

__global__ void edge_den1(const int* __restrict__ ei, const float* __restrict__ s1s,
                          const float* __restrict__ s1d, const float* __restrict__ m1,
                          float* __restrict__ den1) {
  int e = blockIdx.x * blockDim.x + threadIdx.x;
  if (e >= E_TOT) return;
  int src, dst; edge_sd(e, ei, src, dst);
#pragma unroll
  for (int h = 0; h < HEADS; ++h) {
    float v = lrelu(s1s[src * HEADS + h] + s1d[dst * HEADS + h]);
    atomicAdd(&den1[dst * HEADS + h], __expf(v - m1[dst * HEADS + h]));
  }
}

// One wave32 per edge; lane handles 4 consecutive feature dims (head = lane>>3).
__global__ void msg1(const int* __restrict__ ei, const float* __restrict__ h1,
                     const float* __restrict__ s1s, const float* __restrict__ s1d,
                     const float* __restrict__ m1, const float* __restrict__ den1,
                     float* __restrict__ acc) {
  int gid = blockIdx.x * blockDim.x + threadIdx.x;
  int e = gid >> 5;
  int lane = gid & 31;
  if (e >= E_TOT) return;                  // wave-uniform
  int src, dst; edge_sd(e, ei, src, dst);
  int h = lane >> 3;
  float v = lrelu(s1s[src * HEADS + h] + s1d[dst * HEADS + h]);
  float alpha = __expf(v - m1[dst * HEADS + h]) / den1[dst * HEADS + h];
  const float4 hv = *(const float4*)(h1 + (size_t)src * HIDTOT + lane * 4);
  float* out = acc + (size_t)dst * HIDTOT + lane * 4;
  atomicAdd(out + 0, hv.x * alpha);
  atomicAdd(out + 1, hv.y * alpha);
  atomicAdd(out + 2, hv.z * alpha);
  atomicAdd(out + 3, hv.w * alpha);
}

__global__ void finalize1(const float* __restrict__ acc1, const float* __restrict__ bias1,
                          float* __restrict__ h1e) {
  int i = blockIdx.x * blockDim.x + threadIdx.x;
  if (i >= N_NODES * HIDTOT) return;
  float v = acc1[i] + bias1[i & (HIDTOT - 1)];
  h1e[i] = v > 0.0f ? v : expm1f(v);       // ELU
}

__global__ void s2_kernel(const float* __restrict__ h2, const float* __restrict__ as2,
                          const float* __restrict__ ad2,
                          float* __restrict__ s2s, float* __restrict__ s2d) {
  int n = blockIdx.x * blockDim.x + threadIdx.x;
  if (n >= N_NODES) return;
  const float* hp = h2 + (size_t)n * HID;
  float ss = 0.f, sd = 0.f;
#pragma unroll
  for (int d = 0; d < HID; ++d) { float v = hp[d]; ss += v * as2[d]; sd += v * ad2[d]; }
  s2s[n] = ss; s2d[n] = sd;
}

__global__ void edge_max2(const int* __restrict__ ei, const float* __restrict__ s2s,
                          const float* __restrict__ s2d, float* __restrict__ m2) {
  int e = blockIdx.x * blockDim.x + threadIdx.x;
  if (e >= E_TOT) return;
  int src, dst; edge_sd(e, ei, src, dst);
  atomicMaxF32(&m2[dst], lrelu(s2s[src] + s2d[dst]));
}

__global__ void edge_den2(const int* __restrict__ ei, const float* __restrict__ s2s,
                          const float* __restrict__ s2d, const float* __restrict__ m2,
                          float* __restrict__ den2) {
  int e = blockIdx.x * blockDim.x + threadIdx.x;
  if (e >= E_TOT) return;
  int src, dst; edge_sd(e, ei, src, dst);
  atomicAdd(&den2[dst], __expf(lrelu(s2s[src] + s2d[dst]) - m2[dst]));
}

__global__ void msg2(const int* __restrict__ ei, const float* __restrict__ h2,
                     const float* __restrict__ s2s, const float* __restrict__ s2d,
                     const float* __restrict__ m2, const float* __restrict__ den2,
                     float* __restrict__ out) {
  int gid = blockIdx.x * blockDim.x + threadIdx.x;
  int e = gid >> 5;
  int lane = gid & 31;
  if (e >= E_TOT) return;                  // wave-uniform
  int src, dst; edge_sd(e, ei, src, dst);
  float alpha = __expf(lrelu(s2s[src] + s2d[dst]) - m2[dst]) / den2[dst];
  atomicAdd(&out[(size_t)dst * HID + lane], h2[(size_t)src * HID + lane] * alpha);
}

__global__ void bias2_add(float* __restrict__ out, const float* __restrict__ b2) {
  int i = blockIdx.x * blockDim.x + threadIdx.x;
  if (i >= N_NODES * HID) return;
  out[i] += b2[i & (HID - 1)];
}

static inline int cdiv(int a, int b) { return (a + b - 1) / b; }

extern "C" void kernel_launch(void* const* d_in, const int* in_sizes, int n_in,
                              void* d_out, int out_size, void* d_ws, size_t ws_size,
                              hipStream_t stream) {
  const float* x   = (const float*)d_in[0];
  const int*   ei  = (const int*)d_in[1];
  const float* W1  = (const float*)d_in[2];
  const float* as1 = (const float*)d_in[3];
  const float* ad1 = (const float*)d_in[4];
  const float* b1  = (const float*)d_in[5];
  const float* W2  = (const float*)d_in[6];
  const float* as2 = (const float*)d_in[7];
  const float* ad2 = (const float*)d_in[8];
  const float* b2  = (const float*)d_in[9];
  float* out = (float*)d_out;

  float* w    = (float*)d_ws;
  float* bufA = w;                                   // h1, later h1_elu  (N*128)
  float* bufB = bufA + (size_t)N_NODES * HIDTOT;     // acc1             (N*128)
  float* s1s  = bufB + (size_t)N_NODES * HIDTOT;     // N*4
  float* s1d  = s1s + (size_t)N_NODES * HEADS;
  float* m1   = s1d + (size_t)N_NODES * HEADS;
  float* den1 = m1  + (size_t)N_NODES * HEADS;
  float* h2   = den1 + (size_t)N_NODES * HEADS;      // N*32
  float* s2s  = h2  + (size_t)N_NODES * HID;
  float* s2d  = s2s + (size_t)N_NODES;
  float* m2   = s2d + (size_t)N_NODES;
  float* den2 = m2  + (size_t)N_NODES;

  // ---- Layer 1 ----
  // h1 = x @ W1   (100000x128 @ 128x128): 6250*8 = 50000 waves, 4 waves/block
  gemm_f32_wmma<<<12500, 128, 0, stream>>>(x, W1, bufA, N_NODES, HIDTOT, IN_CH);
  s1_kernel<<<cdiv(N_NODES * HEADS, 256), 256, 0, stream>>>(bufA, as1, ad1, s1s, s1d);

  fill_f32<<<cdiv(N_NODES * HEADS, 256), 256, 0, stream>>>(m1, -INFINITY, N_NODES * HEADS);
  fill_f32<<<cdiv(N_NODES * HEADS, 256), 256, 0, stream>>>(den1, 0.0f, N_NODES * HEADS);
  fill_f32<<<cdiv(N_NODES * HIDTOT, 256), 256, 0, stream>>>(bufB, 0.0f, N_NODES * HIDTOT);

  edge_max1<<<cdiv(E_TOT, 256), 256, 0, stream>>>(ei, s1s, s1d, m1);
  edge_den1<<<cdiv(E_TOT, 256), 256, 0, stream>>>(ei, s1s, s1d, m1, den1);
  msg1<<<cdiv(E_TOT * 4, 32), 256, 0, stream>>>(ei, bufA, s1s, s1d, m1, den1, bufB);
  // h1_elu = elu(acc1 + bias1), overwrite bufA (h1 is dead now)
  finalize1<<<cdiv(N_NODES * HIDTOT, 256), 256, 0, stream>>>(bufB, b1, bufA);

  // ---- Layer 2 ----
  // h2 = h1_elu @ W2  (100000x128 @ 128x32): 6250*2 = 12500 waves
  gemm_f32_wmma<<<3125, 128, 0, stream>>>(bufA, W2, h2, N_NODES, HID, HIDTOT);
  s2_kernel<<<cdiv(N_NODES, 256), 256, 0, stream>>>(h2, as2, ad2, s2s, s2d);

  fill_f32<<<cdiv(N_NODES, 256), 256, 0, stream>>>(m2, -INFINITY, N_NODES);
  fill_f32<<<cdiv(N_NODES, 256), 256, 0, stream>>>(den2, 0.0f, N_NODES);
  fill_f32<<<cdiv(N_NODES * HID, 256), 256, 0, stream>>>(out, 0.0f, N_NODES * HID);

  edge_max2<<<cdiv(E_TOT, 256), 256, 0, stream>>>(ei, s2s, s2d, m2);
  edge_den2<<<cdiv(E_TOT, 256), 256, 0, stream>>>(ei, s2s, s2d, m2, den2);
  msg2<<<cdiv(E_TOT * 4, 32), 256, 0, stream>>>(ei, h2, s2s, s2d, m2, den2, out);
  bias2_add<<<cdiv(N_NODES * HID, 256), 256, 0, stream>>>(out, b2);
}